// S2Layer_13984413516035
// MI455X (gfx1250) — compile-verified
//
#include <hip/hip_runtime.h>
#include <hip/hip_bf16.h>

#define N_NODES 100000
#define N_EDGES 1600000
#define IN_DIM  128
#define VAL_DIM 64
#define HID_DIM 128
#define SEMB    8
#define XDIM    136      // 2*VAL_DIM + SEMB
#define GAMMA   1.0f
#define LAM     0.05f
#define ETILES  25       // edge tiles per wave -> 4*25*16 = 1600 edges / block

typedef float v2f __attribute__((ext_vector_type(2)));
typedef float v8f __attribute__((ext_vector_type(8)));

// fp32 WMMA: D(16x16) += A(16x4) * B(4x16), matches reference fp32 numerics.
static __device__ __forceinline__ v8f wmma4(v2f a, v2f b, v8f c) {
  return __builtin_amdgcn_wmma_f32_16x16x4_f32(false, a, false, b, (short)0, c,
                                               false, false);
}

// CDNA5 async DMA: global -> LDS, 16B per lane, tracked with ASYNCcnt.
// LDS aperture rule: LDS byte address = generic shared address [31:0].
static __device__ __forceinline__ void async_g2l_b128(void* lds, const void* g) {
  asm volatile("global_load_async_to_lds_b128 %0, %1, off"
               :: "v"((unsigned)(unsigned long long)(uintptr_t)lds),
                  "v"((unsigned long long)(uintptr_t)g)
               : "memory");
}
static __device__ __forceinline__ void wait_async0() {
  asm volatile("s_wait_asynccnt 0x0" ::: "memory");
}
// Async ops complete in order: waiting <=18 after issuing 18 new ops
// guarantees the previous tile's 18 ops have landed in LDS.
static __device__ __forceinline__ void wait_async18() {
  asm volatile("s_wait_asynccnt 0x12" ::: "memory");
}
static __device__ __forceinline__ void wait_ds0() {
  asm volatile("s_wait_dscnt 0x0" ::: "memory");
}

// ---------------------------------------------------------------------------
// Kernel 1: V = H @ W_v.T ; T = H @ W_t.T   (N x 128) x (128 x 64) each.
// 8 waves / block, one 16-node tile per wave, weights in 64KB LDS panel [n][k].
// ---------------------------------------------------------------------------
__global__ __launch_bounds__(256) void k_node_vt(const float* __restrict__ H,
                                                 const float* __restrict__ Wv,
                                                 const float* __restrict__ Wt,
                                                 float* __restrict__ V,
                                                 float* __restrict__ T) {
  __shared__ __align__(16) float Bvt[128 * 128];  // [n][k], n<64 -> Wv, else Wt
  const int tid = threadIdx.x;
  for (int i = tid; i < 64 * 128; i += 256) {
    Bvt[i] = Wv[i];
    Bvt[64 * 128 + i] = Wt[i];
  }
  __syncthreads();

  const int wave = tid >> 5, lane = tid & 31;
  const int tile = blockIdx.x * 8 + wave;
  const int base = tile * 16;
  if (base >= N_NODES) return;

  const int m = lane & 15, kh = lane >> 4;
  int node = base + m;
  if (node >= N_NODES) node = N_NODES - 1;

  v2f a[32];
  const float* hrow = H + node * 128 + 2 * kh;
#pragma unroll
  for (int ks = 0; ks < 32; ++ks) a[ks] = *(const v2f*)(hrow + 4 * ks);

#pragma unroll
  for (int nt = 0; nt < 8; ++nt) {
    v8f acc = {};
    const float* bp = Bvt + (nt * 16 + m) * 128 + 2 * kh;
#pragma unroll
    for (int ks = 0; ks < 32; ++ks) {
      v2f b = *(const v2f*)(bp + 4 * ks);
      acc = wmma4(a[ks], b, acc);
    }
    const int ng = nt * 16 + m;  // output column of this lane
    float* dst = (nt < 4) ? (V + ng) : (T + (ng - 64));
#pragma unroll
    for (int r = 0; r < 8; ++r) {
      const int nrow = base + r + 8 * kh;
      if (nrow < N_NODES) dst[nrow * 64] = acc[r];
    }
  }
}

// ---------------------------------------------------------------------------
// Kernel 2: edge MLP + softshrink + signed scatter-add + L1 loss.
// 4 waves / block; each wave loops over ETILES tiles of 16 edges so the
// 68KB mlp1_w LDS panel is amortized over 1600 edges. Gathers use CDNA5
// GLOBAL_LOAD_ASYNC_TO_LDS_B128 (ASYNCcnt) with a double-buffered X tile so
// tile t+1's DMA overlaps tile t's 272 WMMAs. All syncs are wave-local.
// ---------------------------------------------------------------------------
static __device__ __forceinline__ void gather_tile(
    const float* __restrict__ V, const float* __restrict__ T,
    const float* __restrict__ semb, const int* __restrict__ erow,
    const int* __restrict__ ecol, const int* __restrict__ esign,
    int e0, int gm, int half, float* X) {
  const int e = e0 + gm;
  const int c = ecol[e], r = erow[e];
  const float* tp = T + c * 64 + half * 32;
  const float* vp = V + r * 64 + half * 32;
  float* xt = X + gm * XDIM + half * 32;
  float* xv = xt + 64;
#pragma unroll
  for (int q = 0; q < 8; ++q) {
    async_g2l_b128(xt + q * 4, tp + q * 4);   // T[col] -> X[m][0:64]
    async_g2l_b128(xv + q * 4, vp + q * 4);   // V[row] -> X[m][64:128]
  }
  if (half == 0) {
    int s = esign[e] + 1;
    s = s < 0 ? 0 : (s > 2 ? 2 : s);
    const float* sp = semb + s * SEMB;
    float* xs = X + gm * XDIM + 128;
    async_g2l_b128(xs, sp);                   // sign_emb -> X[m][128:136]
    async_g2l_b128(xs + 4, sp + 4);
  }
}

__global__ __launch_bounds__(128) void k_edge(
    const float* __restrict__ V, const float* __restrict__ T,
    const float* __restrict__ semb, const float* __restrict__ w1,
    const float* __restrict__ b1, const float* __restrict__ w2,
    const float* __restrict__ b2, const int* __restrict__ erow,
    const int* __restrict__ ecol, const int* __restrict__ esign,
    float* __restrict__ agg, float* __restrict__ lossAcc) {
  __shared__ __align__(16) float W1s[HID_DIM * XDIM];   // [n][k] == mlp1_w
  __shared__ __align__(16) float Xs[4 * 2 * 16 * XDIM]; // per-wave double buffer
  __shared__ __align__(16) float Wl[4 * 16];            // per-wave edge weights

  const int tid = threadIdx.x;
  for (int i = tid; i < HID_DIM * XDIM; i += 128) W1s[i] = w1[i];
  __syncthreads();  // only block-wide sync: W1s ready

  const int wave = tid >> 5, lane = tid & 31;
  const int m = lane & 15, kh = lane >> 4;
  const int gm = lane >> 1, half = lane & 1;  // gather role: edge, 32-col half
  float* X0 = Xs + wave * (2 * 16 * XDIM);
  float* X1 = X0 + 16 * XDIM;
  const int waveBase = blockIdx.x * (4 * ETILES * 16) + wave * (ETILES * 16);
  const float alpha_b = b2[0];
  float lsum = 0.f;

  // prologue: DMA tile 0 into buffer 0
  gather_tile(V, T, semb, erow, ecol, esign, waveBase, gm, half, X0);

  for (int t = 0; t < ETILES; ++t) {
    const int e0 = waveBase + t * 16;
    float* X = (t & 1) ? X1 : X0;

    // overlap: DMA tile t+1 into the other buffer, then wait only for tile t
    if (t + 1 < ETILES) {
      gather_tile(V, T, semb, erow, ecol, esign, e0 + 16, gm, half,
                  (t & 1) ? X0 : X1);
      wait_async18();
    } else {
      wait_async0();
    }

    // ---- edge MLP via fp32 WMMA ------------------------------------------
    v2f a[34];
    const float* xr = X + m * XDIM + 2 * kh;
#pragma unroll
    for (int ks = 0; ks < 34; ++ks) a[ks] = *(const v2f*)(xr + 4 * ks);

    v8f alphaAcc = {};
#pragma unroll
    for (int nt = 0; nt < 8; ++nt) {
      v8f acc = {};
      const float* bp = W1s + (nt * 16 + m) * XDIM + 2 * kh;
#pragma unroll
      for (int ks = 0; ks < 34; ++ks) {
        v2f b = *(const v2f*)(bp + 4 * ks);
        acc = wmma4(a[ks], b, acc);
      }
      const int ng = nt * 16 + m;
      const float bias = b1[ng];
      const float wsc = w2[ng];
#pragma unroll
      for (int r = 0; r < 8; ++r) {
        float h = acc[r] + bias;
        h = h > 0.f ? h : 0.f;   // ReLU
        alphaAcc[r] += h * wsc;  // partial of h @ mlp2_w.T
      }
    }
    // reduce alpha partials across the 16 lanes of each half-wave
#pragma unroll
    for (int off = 1; off < 16; off <<= 1) {
#pragma unroll
      for (int r = 0; r < 8; ++r)
        alphaAcc[r] += __shfl_xor(alphaAcc[r], off, 32);
    }

    // lanes 0 / 16 hold alpha for edges 0..7 / 8..15 of the tile
    if (m == 0) {
#pragma unroll
      for (int r = 0; r < 8; ++r) {
        const int em = 8 * kh + r;
        const int e = e0 + em;
        float al = alphaAcc[r] + alpha_b;
        float ss = al > LAM ? al - LAM : (al < -LAM ? al + LAM : 0.f);
        lsum += fabsf(ss);
        const int s = esign[e];
        Wl[wave * 16 + em] =
            (s > 0 ? ss : 0.f) - GAMMA * (s < 0 ? fabsf(ss) : 0.f);
      }
    }
    wait_ds0();  // Wl visible wave-wide (DS is in-order per wave)

    // ---- scatter: agg[col] += w * v_j (v_j lives in the LDS X tile) -------
#pragma unroll 1
    for (int em = 0; em < 16; ++em) {
      const int e = e0 + em;
      const int c = ecol[e];
      const float w = Wl[wave * 16 + em];
      const float v0 = X[em * XDIM + 64 + lane];
      const float v1 = X[em * XDIM + 96 + lane];
      atomicAdd(agg + c * 64 + lane, w * v0);
      atomicAdd(agg + c * 64 + 32 + lane, w * v1);
    }
  }

  if (m == 0) atomicAdd(lossAcc, lsum);  // 2 atomics per wave total
}

// ---------------------------------------------------------------------------
// Kernel 3: H_new = [H | agg] (N x 192) @ [W_self ; W_out]^T + W_out_b + H
// One fused K=192 GEMM; 96KB LDS weight panel [n][k].
// ---------------------------------------------------------------------------
__global__ __launch_bounds__(256) void k_out(
    const float* __restrict__ H, const float* __restrict__ agg,
    const float* __restrict__ Wself, const float* __restrict__ Woutw,
    const float* __restrict__ Woutb, float* __restrict__ out) {
  __shared__ __align__(16) float B2[128 * 192];
  const int tid = threadIdx.x;
  for (int i = tid; i < 128 * 128; i += 256) {
    const int n = i >> 7, k = i & 127;
    B2[n * 192 + k] = Wself[i];
  }
  for (int i = tid; i < 128 * 64; i += 256) {
    const int n = i >> 6, k = i & 63;
    B2[n * 192 + 128 + k] = Woutw[i];
  }
  __syncthreads();

  const int wave = tid >> 5, lane = tid & 31;
  const int tile = blockIdx.x * 8 + wave;
  const int base = tile * 16;
  if (base >= N_NODES) return;

  const int m = lane & 15, kh = lane >> 4;
  int node = base + m;
  if (node >= N_NODES) node = N_NODES - 1;

  v2f a[48];
  const float* hrow = H + node * 128 + 2 * kh;
#pragma unroll
  for (int ks = 0; ks < 32; ++ks) a[ks] = *(const v2f*)(hrow + 4 * ks);
  const float* grow = agg + node * 64 + 2 * kh;
#pragma unroll
  for (int ks = 0; ks < 16; ++ks) a[32 + ks] = *(const v2f*)(grow + 4 * ks);

#pragma unroll
  for (int nt = 0; nt < 8; ++nt) {
    v8f acc = {};
    const float* bp = B2 + (nt * 16 + m) * 192 + 2 * kh;
#pragma unroll
    for (int ks = 0; ks < 48; ++ks) {
      v2f b = *(const v2f*)(bp + 4 * ks);
      acc = wmma4(a[ks], b, acc);
    }
    const int ng = nt * 16 + m;
    const float bias = Woutb[ng];
#pragma unroll
    for (int r = 0; r < 8; ++r) {
      const int nrow = base + r + 8 * kh;
      if (nrow < N_NODES)
        out[nrow * 128 + ng] = acc[r] + bias + H[nrow * 128 + ng];
    }
  }
}

__global__ void k_loss(const float* __restrict__ lossAcc,
                       float* __restrict__ out) {
  if (threadIdx.x == 0 && blockIdx.x == 0)
    out[(size_t)N_NODES * 128] = lossAcc[0] / (float)N_EDGES;
}

// ---------------------------------------------------------------------------
extern "C" void kernel_launch(void* const* d_in, const int* in_sizes, int n_in,
                              void* d_out, int out_size, void* d_ws,
                              size_t ws_size, hipStream_t stream) {
  (void)in_sizes; (void)n_in; (void)out_size; (void)ws_size;
  const float* H     = (const float*)d_in[0];
  const float* Wv    = (const float*)d_in[1];
  const float* Wt    = (const float*)d_in[2];
  const float* semb  = (const float*)d_in[3];
  const float* w1    = (const float*)d_in[4];
  const float* b1    = (const float*)d_in[5];
  const float* w2    = (const float*)d_in[6];
  const float* b2    = (const float*)d_in[7];
  const float* Wself = (const float*)d_in[8];
  const float* Woutw = (const float*)d_in[9];
  const float* Woutb = (const float*)d_in[10];
  const int* eidx    = (const int*)d_in[11];
  const int* esign   = (const int*)d_in[12];
  const int* erow = eidx;             // edge_index[0]
  const int* ecol = eidx + N_EDGES;   // edge_index[1]

  // Workspace layout (floats): agg[N*64] | loss[16] | V[N*64] | T[N*64]
  float* agg     = (float*)d_ws;
  float* lossAcc = agg + (size_t)N_NODES * 64;
  float* V       = lossAcc + 16;
  float* T       = V + (size_t)N_NODES * 64;
  float* out     = (float*)d_out;

  hipMemsetAsync(agg, 0, ((size_t)N_NODES * 64 + 16) * sizeof(float), stream);

  const int nodeBlocks = (N_NODES + 127) / 128;        // 8 waves x 16 nodes
  const int edgeBlocks = N_EDGES / (4 * ETILES * 16);  // 1000 blocks

  k_node_vt<<<nodeBlocks, 256, 0, stream>>>(H, Wv, Wt, V, T);
  k_edge<<<edgeBlocks, 128, 0, stream>>>(V, T, semb, w1, b1, w2, b2, erow, ecol,
                                         esign, agg, lossAcc);
  k_out<<<nodeBlocks, 256, 0, stream>>>(H, agg, Wself, Woutw, Woutb, out);
  k_loss<<<1, 1, 0, stream>>>(lossAcc, out);
}